// net_39204461478865
// MI455X (gfx1250) — compile-verified
//
#include <hip/hip_runtime.h>

// MI455X / gfx1250, wave32.
//
// y = L3( f^100( relu(x*w1+b1) ) ), f(h)=relu(h@W2+b2), h in R^2, N=16.7M.
// Memory floor ~6us @23.3TB/s; the 1.7G serial 2x2 element-iterations are the
// cost. We batch 32 elements per V_WMMA_F32_16X16X4_F32 (exact fp32):
//
//   A (16x4): W2^T at rows 0,1 x K0,K1   (A lanes 0,1)
//             W2^T at rows 8,9 x K2,K3   (A lanes 24,25)
//   B (4x16): column n = h of element n (elems 0-15 -> K0/K1 rows,
//             elems 16-31 -> K2/K3 rows). With the ISA 32-bit B layout
//             (VGPR = K%2, lane = n + 16*(K/2)) this is simply
//             b.x = h0-per-lane, b.y = h1-per-lane.
//   C: rows 0,1,8,9 biases -> c[0]=b0, c[1]=b1 broadcast; rest 0.
//
// Key trick: D rows 0,1 map to VGPRs 0,1 lanes 0-15 and D rows 8,9 map to
// VGPRs 0,1 lanes 16-31 -- so d[0]/d[1] are the new h0/h1 for ALL 32 lanes,
// already in B layout. Inner loop per 32 element-iterations:
//   1 WMMA + 2 v_max  (no ds_swizzle, no cndmask).

typedef __attribute__((ext_vector_type(2))) float v2f;
typedef __attribute__((ext_vector_type(8))) float v8f;

#define GROUPS 4               // independent 32-element chains per wave (ILP)
#define N_ITERS 100

// ReLU that folds to a single v_max_num_f32 (no NaN-canonicalize max(x,x)):
// select semantics give 0 for NaN input, same as maxNum(x, 0).
__device__ __forceinline__ float relu1(float v) { return v > 0.0f ? v : 0.0f; }

__global__ __launch_bounds__(256) void mlp100_wmma_kernel(
    const float* __restrict__ x,
    const float* __restrict__ w1,
    const float* __restrict__ b1,
    const float* __restrict__ w2,
    const float* __restrict__ b2,
    const float* __restrict__ w3,
    const float* __restrict__ b3,
    float* __restrict__ out,
    int N)
{
    const int lane = (int)(threadIdx.x & 31u);
    const int wave = (int)(threadIdx.x >> 5u);
    const long long base =
        (long long)blockIdx.x * (8LL * 32 * GROUPS) + (long long)wave * (32 * GROUPS);

    // Uniform weight loads (wave-uniform -> scalar loads).
    const float w1_0 = w1[0], w1_1 = w1[1];
    const float b1_0 = b1[0], b1_1 = b1[1];
    const float w2_00 = w2[0], w2_01 = w2[1];   // w2[i*2+j], row-major (2,2)
    const float w2_10 = w2[2], w2_11 = w2[3];
    const float b2_0 = b2[0], b2_1 = b2[1];
    const float w3_0 = w3[0], w3_1 = w3[1];
    const float b3_0 = b3[0];

    // A matrix 16x4 fp32.  Layout: lane m (0-15) holds (K0,K1) in (a.x,a.y);
    // lane 16+m holds (K2,K3).  Place W2^T blocks at rows {0,1}xK{0,1} and
    // rows {8,9}xK{2,3}; everything else zero.
    v2f a = {0.0f, 0.0f};
    if (lane == 0 || lane == 24) { a.x = w2_00; a.y = w2_10; }  // out dim 0: w2[.][0]
    if (lane == 1 || lane == 25) { a.x = w2_01; a.y = w2_11; }  // out dim 1: w2[.][1]

    // C matrix: VGPR0 rows M=0 (lanes 0-15) / M=8 (lanes 16-31) -> bias b0;
    //           VGPR1 rows M=1 / M=9 -> bias b1.  Rows 2..7,10..15 unused.
    v8f c = {b2_0, b2_1, 0.0f, 0.0f, 0.0f, 0.0f, 0.0f, 0.0f};

    // Layer 1: h = relu(x*w1 + b1); lane owns element base + g*32 + lane.
    float h0[GROUPS], h1[GROUPS];
#pragma unroll
    for (int g = 0; g < GROUPS; ++g) {
        long long n = base + (long long)g * 32 + lane;
        long long nc = (n < (long long)N) ? n : (long long)(N - 1);  // clamp, keep EXEC full
        float xv = x[nc];
        h0[g] = relu1(fmaf(xv, w1_0, b1_0));
        h1[g] = relu1(fmaf(xv, w1_1, b1_1));
    }

    // 100 shared-weight iterations, 32 elements per WMMA, 4 chains in flight.
    // d[0]/d[1] come back as new-h0/new-h1 for all 32 lanes in B layout.
#pragma unroll 1
    for (int it = 0; it < N_ITERS; ++it) {
#pragma unroll
        for (int g = 0; g < GROUPS; ++g) {
            v2f b;
            b.x = h0[g];   // B VGPR0: K=0 (lanes 0-15), K=2 (lanes 16-31)
            b.y = h1[g];   // B VGPR1: K=1 (lanes 0-15), K=3 (lanes 16-31)
            v8f d = __builtin_amdgcn_wmma_f32_16x16x4_f32(
                false, a, false, b, (short)0, c, false, false);
            h0[g] = relu1(d[0]);
            h1[g] = relu1(d[1]);
        }
    }

    // Final layer: y = h0*w3[0] + h1*w3[1] + b3
#pragma unroll
    for (int g = 0; g < GROUPS; ++g) {
        long long n = base + (long long)g * 32 + lane;
        float y = fmaf(h1[g], w3_1, fmaf(h0[g], w3_0, b3_0));
        if (n < (long long)N) out[n] = y;
    }
}

extern "C" void kernel_launch(void* const* d_in, const int* in_sizes, int n_in,
                              void* d_out, int out_size, void* d_ws, size_t ws_size,
                              hipStream_t stream) {
    const float* x  = (const float*)d_in[0];
    const float* w1 = (const float*)d_in[1];
    const float* b1 = (const float*)d_in[2];
    const float* w2 = (const float*)d_in[3];
    const float* b2 = (const float*)d_in[4];
    const float* w3 = (const float*)d_in[5];
    const float* b3 = (const float*)d_in[6];
    float* out = (float*)d_out;
    const int N = in_sizes[0];

    const int elems_per_block = 8 * 32 * GROUPS;   // 8 waves * 32 lanes * GROUPS
    const int grid = (N + elems_per_block - 1) / elems_per_block;
    mlp100_wmma_kernel<<<grid, 256, 0, stream>>>(x, w1, b1, w2, b2, w3, b3, out, N);
}